// MultiDimensionalRNN_22239340659386
// MI455X (gfx1250) — compile-verified
//
#include <hip/hip_runtime.h>
#include <hip/hip_bf16.h>
#include <cmath>

// Problem dimensions (match reference).
#define HH     2048
#define WW     2048
#define WSKEW  (HH + WW - 1)   // 4095
#define FC_IN  1024
#define FC_OUT 10
#define MROWS  ((HH * WW) / FC_IN)  // 4096

typedef float v2f __attribute__((ext_vector_type(2)));
typedef float v8f __attribute__((ext_vector_type(8)));

// Pointer types for the async global->LDS copy builtin (signature confirmed
// by the round-2 diagnostic: (v4i AS1*, v4i AS3*, imm offset, imm cpol)).
typedef int v4i __attribute__((vector_size(16)));
typedef __attribute__((address_space(1))) v4i* gv4i_p;
typedef __attribute__((address_space(3))) v4i* lv4i_p;

// gfx1250 has a native V_TANH_F32 (TRANS op per CDNA5 ISA). Prefer the
// builtin (compiler handles TRANS hazards); fall back to inline asm with a
// trailing v_nop to satisfy the 1-op TRANS result hazard rule.
__device__ __forceinline__ float fast_tanh(float x) {
#if __has_builtin(__builtin_amdgcn_tanhf)
    return __builtin_amdgcn_tanhf(x);
#elif __has_builtin(__builtin_amdgcn_tanh_f32)
    return __builtin_amdgcn_tanh_f32(x);
#else
    float r;
    asm("v_tanh_f32 %0, %1\n\tv_nop" : "=v"(r) : "v"(x));
    return r;
#endif
}

// Split workgroup barrier (CDNA5 S_BARRIER_SIGNAL/_WAIT, barrier id -1).
// "arrive" releases this wave's LDS writes and signals; "depart" waits for
// all waves then acquires. Work placed between the two overlaps with other
// waves' arrival — keeps global stores/loads off the serial critical path.
__device__ __forceinline__ void wg_barrier_arrive() {
#if __has_builtin(__builtin_amdgcn_s_barrier_signal)
    __builtin_amdgcn_fence(__ATOMIC_RELEASE, "workgroup", "local");
    __builtin_amdgcn_s_barrier_signal(-1);
#else
    asm volatile("s_wait_dscnt 0x0\n\ts_barrier_signal -1" ::: "memory");
#endif
}
__device__ __forceinline__ void wg_barrier_depart() {
#if __has_builtin(__builtin_amdgcn_s_barrier_wait)
    __builtin_amdgcn_s_barrier_wait(-1);
    __builtin_amdgcn_fence(__ATOMIC_ACQUIRE, "workgroup", "local");
#else
    asm volatile("s_barrier_wait -1" ::: "memory");
#endif
}

// Clamped, always-executed image fetch (uniform EXEC, no divergence):
// returns img[row][idx] inside [0,WW), else 0 (matches skew zero-fill).
__device__ __forceinline__ float ldimg(const float* __restrict__ row, int idx) {
    int cl = idx < 0 ? 0 : (idx > (WW - 1) ? (WW - 1) : idx);
    float v = row[cl];
    return ((unsigned)idx < (unsigned)WW) ? v : 0.0f;
}

// ---------------------------------------------------------------------------
// Kernel 1: skew-domain diagonal scan. One workgroup, 1024 threads (32 waves
// on one WGP), 2 rows/thread. State double-buffered in LDS with a +1 pad so
// the r==0 boundary is a permanent zero cell. Image values are register-
// prefetched 8 steps ahead; all global traffic is issued between the split
// barrier's signal and wait so the 4095-step serial chain only pays
// ds_load -> fma -> v_tanh -> ds_store -> split-barrier per step.
// ---------------------------------------------------------------------------
#define QD 8
__global__ __launch_bounds__(1024) void mdrnn_scan_kernel(
    const float* __restrict__ img,      // [2048*2048]
    const float* __restrict__ w_in,     // [1]
    const float* __restrict__ b_in,     // [1]
    const float* __restrict__ w_state,  // [2] = {k0, k1}
    const float* __restrict__ b_state,  // [1]
    float* __restrict__ acts)           // [2048*2048] flat activations out
{
    __shared__ float hbuf[2][HH + 1];   // hbuf[p][r+1] = state of row r

    const int   t  = threadIdx.x;
    const float w  = w_in[0];
    const float b  = b_in[0];
    const float k0 = w_state[0];
    const float k1 = w_state[1];
    const float bb = b_state[0] + b;    // bs + b_in folded together

    const int r0 = t;
    const int r1 = t + 1024;

    hbuf[0][r0 + 1] = 0.0f;
    hbuf[0][r1 + 1] = 0.0f;
    if (t == 0) { hbuf[0][0] = 0.0f; hbuf[1][0] = 0.0f; }  // permanent r=-1 cell
    __syncthreads();

    const float* __restrict__ row0 = img + (size_t)r0 * WW;
    const float* __restrict__ row1 = img + (size_t)r1 * WW;

    // Depth-QD register prefetch queues along each thread's image row.
    float qa[QD], qb[QD];
#pragma unroll
    for (int j = 0; j < QD; ++j) {
        qa[j] = ldimg(row0, j - r0);
        qb[j] = ldimg(row1, j - r1);
    }

    int p = 0;
#pragma unroll 8
    for (int c = 0; c < WSKEW; ++c) {
        const int q = p ^ 1;

        // Neighbor (r-1) and self states: adjacent pair, branch-free.
        const float hm0 = hbuf[p][r0];
        const float hc0 = hbuf[p][r0 + 1];
        const float hm1 = hbuf[p][r1];
        const float hc1 = hbuf[p][r1 + 1];

        const float n0 = fast_tanh(fmaf(k0, hm0, fmaf(k1, hc0, fmaf(w, qa[0], bb))));
        const float n1 = fast_tanh(fmaf(k0, hm1, fmaf(k1, hc1, fmaf(w, qb[0], bb))));

        hbuf[q][r0 + 1] = n0;
        hbuf[q][r1 + 1] = n1;

        // Signal arrival (after draining our LDS writes only), then do the
        // off-critical-path work while the other 31 waves arrive.
        wg_barrier_arrive();

        const int c0 = c - r0;
        const int c1 = c - r1;
        if ((unsigned)c0 < (unsigned)WW) acts[(size_t)r0 * WW + c0] = n0;
        if ((unsigned)c1 < (unsigned)WW) acts[(size_t)r1 * WW + c1] = n1;

        // Rotate queues; issue the load QD steps ahead of use.
#pragma unroll
        for (int j = 0; j < QD - 1; ++j) { qa[j] = qa[j + 1]; qb[j] = qb[j + 1]; }
        qa[QD - 1] = ldimg(row0, c + QD - r0);
        qb[QD - 1] = ldimg(row1, c + QD - r1);

        p = q;
        wg_barrier_depart();
    }
}

// ---------------------------------------------------------------------------
// Kernel 2: out = acts[4096x1024] @ fc_w[10x1024]^T + fc_b via
// V_WMMA_F32_16X16X4_F32. fc_w is staged once into LDS (zero-padded to 16
// rows, stride-padded against bank conflicts; async-to-LDS copy), and the
// K loop is register double-buffered so A/B loads for group g+1 are in
// flight under the WMMA chain of group g.
//
// VGPR layouts per CDNA5 ISA 7.12.2 (wave32):
//   A 16x4 : lanes 0-15 -> M=lane, {v0=K0, v1=K1}; lanes 16-31 -> {K2, K3}
//   B 4x16 : column-major mirror: lane&15 = N, halves split K the same way
//   C/D    : v8f, element i -> M = i + 8*(lane>=16), N = lane&15
// ---------------------------------------------------------------------------
#define BSTRIDE 1028   // floats; 4112B row stride: 16B aligned, banks spread
__global__ __launch_bounds__(256) void mdrnn_fc_wmma_kernel(
    const float* __restrict__ acts,   // [4096 x 1024]
    const float* __restrict__ fc_w,   // [10 x 1024]
    const float* __restrict__ fc_b,   // [10]
    float* __restrict__ out)          // [4096 x 10]
{
    __shared__ float bsh[16 * BSTRIDE];

    const int tid  = threadIdx.x;
    const int lane = tid & 31;
    const int wave = tid >> 5;
    const int l    = lane & 15;       // N (and A-row offset)
    const int half = lane >> 4;       // K-half selector
    const int m0   = (blockIdx.x * 8 + wave) * 16;

    // Zero pad rows 10..15 (N padding -> no masking in the inner loop).
    for (int i = tid; i < 6 * FC_IN; i += 256) {
        bsh[(FC_OUT + i / FC_IN) * BSTRIDE + (i % FC_IN)] = 0.0f;
    }
    // Stage fc_w rows 0..9 into LDS, 16B per thread per op.
#if __has_builtin(__builtin_amdgcn_global_load_async_to_lds_b128)
    for (int i = tid; i < (FC_OUT * FC_IN) / 4; i += 256) {
        const int rr = (i * 4) / FC_IN;
        const int cc = (i * 4) % FC_IN;
        __builtin_amdgcn_global_load_async_to_lds_b128(
            (gv4i_p)(fc_w + rr * FC_IN + cc),
            (lv4i_p)&bsh[rr * BSTRIDE + cc],
            0, 0);
    }
#if __has_builtin(__builtin_amdgcn_s_wait_asynccnt)
    __builtin_amdgcn_s_wait_asynccnt(0);
#else
    asm volatile("s_wait_asynccnt 0x0" ::: "memory");
#endif
#else
    for (int i = tid; i < (FC_OUT * FC_IN) / 4; i += 256) {
        const int rr = (i * 4) / FC_IN;
        const int cc = (i * 4) % FC_IN;
        const float4 v = *(const float4*)(fc_w + rr * FC_IN + cc);
        *(float4*)&bsh[rr * BSTRIDE + cc] = v;
    }
#endif
    __syncthreads();

    const float* __restrict__ arow = acts + (size_t)(m0 + l) * FC_IN;
    const float* __restrict__ blds = &bsh[l * BSTRIDE];

    // Register double-buffered K loop: groups of 4 WMMA steps (16 K).
    v2f a[4], bv[4], an[4], bn[4];
#define LDG(aa, bb, kbase)                                        \
    _Pragma("unroll") for (int u = 0; u < 4; ++u) {               \
        const int kk = (kbase) + 4 * u + 2 * half;                \
        (aa)[u].x = arow[kk]; (aa)[u].y = arow[kk + 1];           \
        (bb)[u].x = blds[kk]; (bb)[u].y = blds[kk + 1];           \
    }

    LDG(a, bv, 0)
    v8f c = {};
    for (int k = 0; k < FC_IN; k += 16) {
        const bool more = (k + 16) < FC_IN;
        if (more) { LDG(an, bn, k + 16) }
#pragma unroll
        for (int u = 0; u < 4; ++u) {
            // 8 args: (neg_a, A, neg_b, B, c_mod, C, reuse_a, reuse_b)
            c = __builtin_amdgcn_wmma_f32_16x16x4_f32(
                    false, a[u], false, bv[u], (short)0, c, false, false);
        }
        if (more) {
#pragma unroll
            for (int u = 0; u < 4; ++u) { a[u] = an[u]; bv[u] = bn[u]; }
        }
    }
#undef LDG

    const float bias = (l < FC_OUT) ? fc_b[l] : 0.0f;
#pragma unroll
    for (int i = 0; i < 8; ++i) {
        const int m = m0 + i + 8 * half;
        if (l < FC_OUT)
            out[(size_t)m * FC_OUT + l] = c[i] + bias;
    }
}

// ---------------------------------------------------------------------------
extern "C" void kernel_launch(void* const* d_in, const int* in_sizes, int n_in,
                              void* d_out, int out_size, void* d_ws, size_t ws_size,
                              hipStream_t stream) {
    (void)in_sizes; (void)n_in; (void)out_size; (void)ws_size;

    const float* x       = (const float*)d_in[0];
    const float* w_in    = (const float*)d_in[1];
    const float* b_in    = (const float*)d_in[2];
    const float* w_state = (const float*)d_in[3];
    const float* b_state = (const float*)d_in[4];
    const float* fc_w    = (const float*)d_in[5];
    const float* fc_b    = (const float*)d_in[6];

    float* acts = (float*)d_ws;  // 2048*2048 floats = 16 MB scratch

    mdrnn_scan_kernel<<<1, 1024, 0, stream>>>(x, w_in, b_in, w_state, b_state, acts);

    // 4096 M-rows / (8 waves * 16 rows) = 32 blocks of 256 threads.
    mdrnn_fc_wmma_kernel<<<32, 256, 0, stream>>>(acts, fc_w, fc_b, (float*)d_out);
}